// OnLane_38019050504608
// MI455X (gfx1250) — compile-verified
//
#include <hip/hip_runtime.h>
#include <math.h>

// Reference shapes: A=256, T=64 -> N=16384 queries; M=8192 centerline points.
// dist^2 via rank-4 WMMA:  A[m,:]=[-2qx,-2qy,|q|^2,1], B[:,n]=[cx,cy,1,|c|^2]
// angle mask via rank-2 WMMA dot: qdir.cdir > |qdir|*cos(0.2)  (exact for diff in [0,pi])
// dist<5 mask deferred: min over angle-ok d2, then (min<25 ? sqrt(min) : inf) per row.

typedef __attribute__((ext_vector_type(2))) float v2f;
typedef __attribute__((ext_vector_type(4))) float v4f;
typedef __attribute__((ext_vector_type(8))) float v8f;

#define T_STEPS 64
#define COS_0P2 0.98006657784124163f   // cos(0.2)

// Phase 0a: per-query step direction, |dir|*cos(0.2) threshold, init mins to +inf.
__global__ void prep_q_kernel(const float* __restrict__ xy,
                              float* __restrict__ qdir,
                              float* __restrict__ qncos,
                              float* __restrict__ min_ws,
                              int N) {
    int i = blockIdx.x * blockDim.x + threadIdx.x;
    if (i >= N) return;
    int t = i % T_STEPS;
    float dx, dy;
    if (t < T_STEPS - 1) {
        dx = xy[2*(i+1)]   - xy[2*i];
        dy = xy[2*(i+1)+1] - xy[2*i+1];
    } else {  // last step repeats final segment
        dx = xy[2*i]   - xy[2*(i-1)];
        dy = xy[2*i+1] - xy[2*(i-1)+1];
    }
    qdir[2*i]   = dx;
    qdir[2*i+1] = dy;
    qncos[i]    = sqrtf(dx*dx + dy*dy) * COS_0P2;
    min_ws[i]   = __builtin_inff();
}

// Phase 0b: pack per-(tile,lane) WMMA B operands: [Bd0, Bd1, Ba0, Ba1].
// B layout (16x4, K in rows): VGPR0 = K0 (lanes 0-15) / K2 (lanes 16-31),
//                             VGPR1 = K1 / K3.
__global__ void pack_c_kernel(const float* __restrict__ cxy,
                              const float* __restrict__ cdir,
                              v4f* __restrict__ bpack,
                              int M) {
    int j = blockIdx.x * blockDim.x + threadIdx.x;   // (tile<<5) | lane
    if (j >= 2 * M) return;
    int lane = j & 31;
    bool hi  = lane >= 16;
    int ci   = ((j >> 5) << 4) + (lane & 15);
    float cx  = cxy[2*ci],  cy  = cxy[2*ci+1];
    float cdx = cdir[2*ci], cdy = cdir[2*ci+1];
    v4f p;
    p.x = hi ? 1.0f            : cx;    // Bd: K2 : K0
    p.y = hi ? (cx*cx + cy*cy) : cy;    // Bd: K3 : K1
    p.z = hi ? 0.0f            : cdx;   // Ba: K2 : K0
    p.w = hi ? 0.0f            : cdy;   // Ba: K3 : K1
    bpack[j] = p;
}

// Phase 1: one wave per 16-query tile; sweep all M/16 centerline tiles with WMMA.
__global__ void __launch_bounds__(256)
pair_min_kernel(const float* __restrict__ xy,
                const float* __restrict__ qdir,
                const float* __restrict__ qncos,
                const v4f* __restrict__ bpack,
                float* __restrict__ min_ws,
                int N, int M) {
    const int lane  = threadIdx.x & 31;
    const int wave  = (blockIdx.x * blockDim.x + threadIdx.x) >> 5;
    const int qbase = wave * 16;
    if (qbase >= N) return;                    // wave-uniform: EXEC stays all-1s

    const bool hi = lane >= 16;
    const int  qi = qbase + (lane & 15);

    // A operand (dist^2 GEMM): VGPR0 = K0/K2 halves, VGPR1 = K1/K3.
    const float qx = xy[2*qi], qy = xy[2*qi+1];
    v2f Ad;
    Ad.x = hi ? (qx*qx + qy*qy) : (-2.0f * qx);
    Ad.y = hi ? 1.0f            : (-2.0f * qy);

    // A operand (angle-dot GEMM): [qdx, qdy, 0, 0]
    const float qdx = qdir[2*qi], qdy = qdir[2*qi+1];
    v2f Aa;
    Aa.x = hi ? 0.0f : qdx;
    Aa.y = hi ? 0.0f : qdy;

    // Per-row angle thresholds for this lane's D slice: rows r + 8*hi.
    float thr[8];
    #pragma unroll
    for (int r = 0; r < 8; ++r) thr[r] = qncos[qbase + r + (hi ? 8 : 0)];

    const float INF = __builtin_inff();
    float m[8];
    #pragma unroll
    for (int r = 0; r < 8; ++r) m[r] = INF;

    const v4f* bp = bpack + lane;
    const int ntiles = M >> 4;
    #pragma unroll 2
    for (int ct = 0; ct < ntiles; ++ct) {
        const v4f p = bp[ct << 5];                       // one global_load_b128
        __builtin_prefetch(&bp[(ct + 2) << 5], 0, 1);    // global_prefetch_b8

        v2f Bd; Bd.x = p.x; Bd.y = p.y;
        v2f Ba; Ba.x = p.z; Ba.y = p.w;
        v8f Cz = {};
        v8f Dd = __builtin_amdgcn_wmma_f32_16x16x4_f32(false, Ad, false, Bd,
                                                       (short)0, Cz, false, false);
        v8f Da = __builtin_amdgcn_wmma_f32_16x16x4_f32(false, Aa, false, Ba,
                                                       (short)0, Cz, false, false);
        #pragma unroll
        for (int r = 0; r < 8; ++r) {
            float cand = (Da[r] > thr[r]) ? Dd[r] : INF;   // angle mask only
            // raw min (inputs never NaN) -> single v_min_num_f32, no canonicalize
            asm("v_min_num_f32 %0, %1, %2" : "=v"(m[r]) : "v"(m[r]), "v"(cand));
        }
    }

    // Min-reduce across the 16 lanes of each half (xor masks < 16 keep halves apart).
    #pragma unroll
    for (int off = 1; off < 16; off <<= 1) {
        #pragma unroll
        for (int r = 0; r < 8; ++r) {
            float o = __shfl_xor(m[r], off, 32);
            asm("v_min_num_f32 %0, %1, %2" : "=v"(m[r]) : "v"(m[r]), "v"(o));
        }
    }
    if ((lane & 15) == 0) {
        #pragma unroll
        for (int r = 0; r < 8; ++r) {
            // deferred distance mask: empty set (or min>=25) -> inf
            float v = (m[r] < 25.0f) ? sqrtf(m[r]) : INF;
            min_ws[qbase + r + (hi ? 8 : 0)] = v;
        }
    }
}

// Phase 2: mean over N per-query mins.
__global__ void mean_kernel(const float* __restrict__ min_ws,
                            float* __restrict__ out, int N) {
    __shared__ float sm[256];
    float s = 0.0f;
    for (int i = threadIdx.x; i < N; i += blockDim.x) s += min_ws[i];
    sm[threadIdx.x] = s;
    __syncthreads();
    for (int off = 128; off > 0; off >>= 1) {
        if ((int)threadIdx.x < off) sm[threadIdx.x] += sm[threadIdx.x + off];
        __syncthreads();
    }
    if (threadIdx.x == 0) out[0] = sm[0] / (float)N;
}

extern "C" void kernel_launch(void* const* d_in, const int* in_sizes, int n_in,
                              void* d_out, int out_size, void* d_ws, size_t ws_size,
                              hipStream_t stream) {
    const float* xy   = (const float*)d_in[0];   // [A,T,2] flat, N = A*T
    const float* cxy  = (const float*)d_in[1];   // [M,2]
    const float* cdir = (const float*)d_in[2];   // [M,2] unit vectors

    const int N = in_sizes[0] / 2;
    const int M = in_sizes[1] / 2;

    // Workspace (floats): bpack[8M] (16B-aligned, first) | qdir[2N] | qncos[N] | min_ws[N]
    v4f*   bpack  = (v4f*)d_ws;                  // 2M entries of 16B
    float* qdir   = (float*)(bpack + 2 * (size_t)M);
    float* qncos  = qdir + 2 * (size_t)N;
    float* min_ws = qncos + N;

    prep_q_kernel<<<(N + 255) / 256, 256, 0, stream>>>(xy, qdir, qncos, min_ws, N);
    pack_c_kernel<<<(2 * M + 255) / 256, 256, 0, stream>>>(cxy, cdir, bpack, M);

    const int waves   = (N + 15) / 16;
    const int threads = waves * 32;
    pair_min_kernel<<<(threads + 255) / 256, 256, 0, stream>>>(
        xy, qdir, qncos, bpack, min_ws, N, M);

    mean_kernel<<<1, 256, 0, stream>>>(min_ws, (float*)d_out, N);
}